// DotProductAttentionWithRPR_23871428231824
// MI455X (gfx1250) — compile-verified
//
#include <hip/hip_runtime.h>
#include <math.h>

typedef float v2f __attribute__((ext_vector_type(2)));
typedef float v8f __attribute__((ext_vector_type(8)));

#define BH 64
#define S 1024
#define D 64
#define NEGV (-1000000.0f)
#define SCALE 0.125f   // 1/sqrt(64)

__device__ __forceinline__ v2f ldg2(const float* p) { return *(const v2f*)p; }

// ---------------------------------------------------------------------------
// K_A: content scores  ws[b,q,k] = sum_d Q[b,q,d] * K[b,k,d]   (raw, unscaled)
// grid (BH, S/32), 256 threads (8 waves). Each block: 32 q-rows x 1024 k.
// ---------------------------------------------------------------------------
__global__ __launch_bounds__(256) void ka_content_scores(
    const float* __restrict__ Q, const float* __restrict__ K,
    float* __restrict__ ws)
{
  const int b    = blockIdx.x;
  const int qt   = blockIdx.y;        // 32-row q tile
  const int tid  = threadIdx.x;
  const int wave = tid >> 5;
  const int lane = tid & 31;
  const int lr   = lane & 15;
  const int hi   = lane >> 4;         // lane half: K offset 0 or 2

  const float* Qb  = Q + (size_t)b * S * D;
  const float* Kb  = K + (size_t)b * S * D;
  float*       wsb = ws + ((size_t)b * S + (size_t)qt * 32) * S;

  // Preload A fragments for both 16-row m-tiles: 16 steps of K=4 over D=64
  v2f afrag[2][16];
  #pragma unroll
  for (int mt = 0; mt < 2; ++mt) {
    const float* qrow = Qb + (size_t)(qt * 32 + mt * 16 + lr) * D + hi * 2;
    #pragma unroll
    for (int s = 0; s < 16; ++s) afrag[mt][s] = ldg2(qrow + s * 4);
  }

  // 2 m-tiles x 64 k-tiles = 128 output tiles; 16 per wave
  for (int t = wave * 16; t < wave * 16 + 16; ++t) {
    const int mt = t & 1;
    const int kt = (t >> 1) * 16;
    const float* krow = Kb + (size_t)(kt + lr) * D + hi * 2;   // B[d][n]=K[n][d]
    v8f c = {};
    #pragma unroll
    for (int s = 0; s < 16; ++s) {
      v2f bfrag = ldg2(krow + s * 4);
      c = __builtin_amdgcn_wmma_f32_16x16x4_f32(false, afrag[mt][s],
                                                false, bfrag,
                                                (short)0, c, false, false);
    }
    #pragma unroll
    for (int j = 0; j < 8; ++j) {
      const int row = mt * 16 + j + hi * 8;
      wsb[(size_t)row * S + kt + lr] = c[j];
    }
  }
}

// ---------------------------------------------------------------------------
// K_B: per fixed q (grid.x) and 16-batch tile (grid.y):
//   s = (ws_content + Q[:,q,:] @ pos_k[q]^T) * scale, mask, softmax -> LDS
//   waves 0-3: out[b,q,:]  = attn @ pos_v[q]   (plain store, full coverage)
//   waves 4-7: attn -> ws  (for K_C)
// ---------------------------------------------------------------------------
__global__ __launch_bounds__(256) void kb_pos_softmax(
    const float* __restrict__ Q, const float* __restrict__ PK,
    const float* __restrict__ PV, const int* __restrict__ vlen,
    float* __restrict__ ws, float* __restrict__ out)
{
  __shared__ float sm[16 * S];        // 64 KB attn slab: 16 batches x 1024 k
  const int q    = blockIdx.x;
  const int bt   = blockIdx.y;        // batch tile (16 of BH=64)
  const int tid  = threadIdx.x;
  const int wave = tid >> 5;
  const int lane = tid & 31;
  const int lr   = lane & 15;
  const int hi   = lane >> 4;

  // A = Q[bt*16 .. +16][q][:]  (M = batch, K = d)
  v2f afrag[16];
  {
    const float* qp = Q + ((size_t)(bt * 16 + lr) * S + q) * D + hi * 2;
    #pragma unroll
    for (int s = 0; s < 16; ++s) afrag[s] = ldg2(qp + s * 4);
  }

  const float* pkq = PK + (size_t)q * S * D;
  // 64 k-tiles / 8 waves
  for (int ti = 0; ti < 8; ++ti) {
    const int kt = (wave * 8 + ti) * 16;
    const float* bp = pkq + (size_t)(kt + lr) * D + hi * 2;  // B[d][k]=pos_k[q][k][d]
    v8f c = {};
    #pragma unroll
    for (int s = 0; s < 16; ++s) {
      v2f bfrag = ldg2(bp + s * 4);
      c = __builtin_amdgcn_wmma_f32_16x16x4_f32(false, afrag[s],
                                                false, bfrag,
                                                (short)0, c, false, false);
    }
    #pragma unroll
    for (int j = 0; j < 8; ++j) {
      const int bl  = j + hi * 8;                 // local batch row 0..15
      const int b   = bt * 16 + bl;
      const int col = kt + lr;
      float v = (c[j] + ws[((size_t)b * S + q) * S + col]) * SCALE;
      if (col >= vlen[b]) v = NEGV;
      sm[bl * S + col] = v;
    }
  }
  __syncthreads();

  // Row softmax: 16 rows / 8 waves
  for (int r = wave; r < 16; r += 8) {
    float m = -3.0e38f;
    for (int jj = lane; jj < S; jj += 32) m = fmaxf(m, sm[r * S + jj]);
    #pragma unroll
    for (int off = 16; off > 0; off >>= 1) m = fmaxf(m, __shfl_xor(m, off, 32));
    float sum = 0.0f;
    for (int jj = lane; jj < S; jj += 32) {
      float e = __expf(sm[r * S + jj] - m);
      sm[r * S + jj] = e;
      sum += e;
    }
    #pragma unroll
    for (int off = 16; off > 0; off >>= 1) sum += __shfl_xor(sum, off, 32);
    const float inv = 1.0f / sum;
    for (int jj = lane; jj < S; jj += 32) sm[r * S + jj] *= inv;
  }
  __syncthreads();

  if (wave < 4) {
    // out_pos tile: 16 batches x 16 d-cols (ntile = wave), K = 1024
    const int nt = wave * 16;
    const float* pvq = PV + (size_t)q * S * D;
    v8f c = {};
    #pragma unroll 4
    for (int kk = 0; kk < S; kk += 4) {
      v2f a;
      a.x = sm[lr * S + kk + hi * 2];
      a.y = sm[lr * S + kk + hi * 2 + 1];
      const float* bp = pvq + (size_t)(kk + hi * 2) * D + nt + lr;
      v2f bf;
      bf.x = bp[0];                     // pos_v[q][kk+hi*2  ][n]
      bf.y = bp[D];                     // pos_v[q][kk+hi*2+1][n]
      c = __builtin_amdgcn_wmma_f32_16x16x4_f32(false, a, false, bf,
                                                (short)0, c, false, false);
    }
    #pragma unroll
    for (int j = 0; j < 8; ++j) {
      const int b = bt * 16 + j + hi * 8;
      out[((size_t)b * S + q) * D + nt + lr] = c[j];   // single owner per element
    }
  } else {
    // spill attn rows to ws for K_C (4 rows per wave, coalesced)
    for (int r = (wave - 4) * 4; r < (wave - 4) * 4 + 4; ++r) {
      const int b = bt * 16 + r;
      float* dst = &ws[((size_t)b * S + q) * S];
      for (int jj = lane; jj < S; jj += 32) dst[jj] = sm[r * S + jj];
    }
  }
}

// ---------------------------------------------------------------------------
// K_C: content output  out[b,q,:] += attn[b,q,:] @ V[b]
// grid (BH, S/32); wave w: m-tile = w&1, n-tile = w>>1. K = 1024.
// ---------------------------------------------------------------------------
__global__ __launch_bounds__(256) void kc_content_out(
    const float* __restrict__ V, const float* __restrict__ ws,
    float* __restrict__ out)
{
  const int b    = blockIdx.x;
  const int qt   = blockIdx.y;
  const int tid  = threadIdx.x;
  const int wave = tid >> 5;
  const int lane = tid & 31;
  const int lr   = lane & 15;
  const int hi   = lane >> 4;

  const int mt    = wave & 1;
  const int nt    = (wave >> 1) * 16;
  const int qbase = qt * 32 + mt * 16;

  const float* attn = ws + ((size_t)b * S + qbase) * S;
  const float* Vb   = V + (size_t)b * S * D;

  v8f c = {};
  #pragma unroll 4
  for (int kk = 0; kk < S; kk += 4) {
    v2f a = ldg2(attn + (size_t)lr * S + kk + hi * 2);
    const float* bp = Vb + (size_t)(kk + hi * 2) * D + nt + lr;
    v2f bf;
    bf.x = bp[0];
    bf.y = bp[D];
    c = __builtin_amdgcn_wmma_f32_16x16x4_f32(false, a, false, bf,
                                              (short)0, c, false, false);
  }
  #pragma unroll
  for (int j = 0; j < 8; ++j) {
    const int row = qbase + j + hi * 8;
    const size_t idx = ((size_t)b * S + row) * D + nt + lr;
    out[idx] += c[j];                  // pos part already stored by K_B
  }
}

// ---------------------------------------------------------------------------
extern "C" void kernel_launch(void* const* d_in, const int* in_sizes, int n_in,
                              void* d_out, int out_size, void* d_ws, size_t ws_size,
                              hipStream_t stream) {
  const float* Q    = (const float*)d_in[0];
  const float* K    = (const float*)d_in[1];
  const float* V    = (const float*)d_in[2];
  const float* PK   = (const float*)d_in[3];
  const float* PV   = (const float*)d_in[4];
  const int*   vlen = (const int*)d_in[5];
  float* out = (float*)d_out;
  float* ws  = (float*)d_ws;           // needs BH*S*S*4 = 256 MB

  dim3 blk(256);
  ka_content_scores<<<dim3(BH, S / 32), blk, 0, stream>>>(Q, K, ws);
  kb_pos_softmax<<<dim3(S, BH / 16), blk, 0, stream>>>(Q, PK, PV, vlen, ws, out);
  kc_content_out<<<dim3(BH, S / 32), blk, 0, stream>>>(V, ws, out);
}